// PointPillarScatter_62216896250120
// MI455X (gfx1250) — compile-verified
//
#include <hip/hip_runtime.h>
#include <stdint.h>

// ---- problem constants (from reference) ----
#define NUM_FEATURES 64
#define NX 704
#define NY 200
#define MAX_CAV 5
#define NUM_PIXELS (NY * NX)                         // 140800
#define OUT_ELEMS (MAX_CAV * NUM_FEATURES * NUM_PIXELS)  // 45,056,000 floats

// -------------------------------------------------------------------------
// Kernel 1: zero-fill the 180 MB canvas through the CDNA5 async
// LDS -> global store path (GLOBAL_STORE_ASYNC_FROM_LDS_B128, ASYNCcnt).
// Each lane zeroes its own 16B LDS chunk once, then streams fire-and-forget
// 16B async stores over a grid-stride range. One s_wait_asynccnt 0 at end.
// -------------------------------------------------------------------------
__global__ void __launch_bounds__(256)
fill_zero_async(float* __restrict__ out, int n16 /*16B chunks*/, int n_total /*floats*/) {
    __shared__ __align__(16) float zbuf[256 * 4];
    const int tid = threadIdx.x;

    // zero this thread's private 16B LDS chunk
    float4 z;
    z.x = 0.0f; z.y = 0.0f; z.z = 0.0f; z.w = 0.0f;
    *reinterpret_cast<float4*>(&zbuf[tid * 4]) = z;
    __syncthreads();  // LDS zeros visible before any async store reads them

    // LDS byte offset = low 32 bits of the generic shared-space address
    const uint32_t lds_off = (uint32_t)(uintptr_t)(&zbuf[tid * 4]);

    const int gid    = blockIdx.x * blockDim.x + tid;
    const int stride = gridDim.x * blockDim.x;
    const uint64_t base = (uint64_t)(uintptr_t)out;

    for (int i = gid; i < n16; i += stride) {
        uint64_t addr = base + ((uint64_t)i << 4);
        // GV mode: 64-bit vaddr pair, LDS source vgpr, no saddr
        asm volatile("global_store_async_from_lds_b128 %0, %1, off"
                     :: "v"(addr), "v"(lds_off)
                     : "memory");
    }

    // generic tail (OUT_ELEMS is divisible by 4, so normally dead)
    const int tail_base = n16 * 4;
    const int tail      = n_total - tail_base;
    if (gid < tail) out[tail_base + gid] = 0.0f;

    // drain all outstanding async stores before wave exit
    asm volatile("s_wait_asynccnt 0x0" ::: "memory");
}

// -------------------------------------------------------------------------
// Kernel 2: scatter pillar features into the canvas.
// One thread handles 4 consecutive features of one pillar:
//   i = p*16 + q, features c = 4q..4q+3, so the float4 feature load index
//   is exactly i (p*64 + 4q == 4*i) -> perfectly coalesced b128 loads.
// Stores go to out[b*64*NP + c*NP + (y*NX + x)], stride NP apart.
// -------------------------------------------------------------------------
__global__ void __launch_bounds__(256)
scatter_pillars(const int* __restrict__ vc, const float* __restrict__ feat,
                float* __restrict__ out, int n_quads /* npillars*16 */) {
    const int i = blockIdx.x * blockDim.x + threadIdx.x;
    if (i >= n_quads) return;

    const int p  = i >> 4;          // pillar index
    const int c4 = (i & 15) << 2;   // first of 4 feature channels

    // coords: [b, z, y, x]; 16 threads per pillar read the same 16B -> broadcast
    const int b = vc[4 * p + 0];
    const int y = vc[4 * p + 2];
    const int x = vc[4 * p + 3];
    const int rem = y * NX + x;

    const float4 f = reinterpret_cast<const float4*>(feat)[i];

    // all indices < 45,056,000 -> fits 32-bit, but keep size_t for addressing
    const size_t base = (size_t)b * (NUM_FEATURES * (size_t)NUM_PIXELS)
                      + (size_t)c4 * NUM_PIXELS + (size_t)rem;

    out[base                  ] = f.x;
    out[base +     NUM_PIXELS ] = f.y;
    out[base + 2 * NUM_PIXELS ] = f.z;
    out[base + 3 * NUM_PIXELS ] = f.w;
}

// -------------------------------------------------------------------------
// Launcher: fill (async-store path) then scatter, serialized on `stream`.
// -------------------------------------------------------------------------
extern "C" void kernel_launch(void* const* d_in, const int* in_sizes, int n_in,
                              void* d_out, int out_size, void* d_ws, size_t ws_size,
                              hipStream_t stream) {
    const int*   vc   = (const int*)d_in[0];    // voxel_coords [N,4] int32
    const float* feat = (const float*)d_in[1];  // pillar_features [N,64] f32
    float*       out  = (float*)d_out;          // [5,64,200,704] f32

    // ---- pass 1: zero fill ----
    const int n16 = out_size >> 2;              // number of 16B chunks
    const int fill_blocks = 2048;               // grid-stride; ~21 stores/thread
    fill_zero_async<<<fill_blocks, 256, 0, stream>>>(out, n16, out_size);

    // ---- pass 2: scatter ----
    const int npillars = in_sizes[0] / 4;
    const int n_quads  = npillars * (NUM_FEATURES / 4);  // 960,000
    const int blocks   = (n_quads + 255) / 256;
    scatter_pillars<<<blocks, 256, 0, stream>>>(vc, feat, out, n_quads);
}